// NewGraphReasoner_6528350290170
// MI455X (gfx1250) — compile-verified
//
#include <hip/hip_runtime.h>
#include <math.h>

// ---------------- problem constants (match reference) ----------------
constexpr int HDIM   = 256;
constexpr int LNUM   = 4;
constexpr int NNODES = 100000;
constexpr int EDGES  = 300000;
constexpr float BN_EPS  = 1e-5f;
constexpr float AGG_EPS = 1e-6f;

typedef __attribute__((ext_vector_type(2))) float v2f;
typedef __attribute__((ext_vector_type(8))) float v8f;

// GEMM tiling: 256 threads = 8 waves; block tile 32(M) x 64(N), K-chunk 16.
// Each wave owns one 16x16 WMMA tile: waves laid out 2(M) x 4(N).
#define TB_M 32
#define TB_N 64
#define TB_K 16
#define A_PAD 2   // row stride 18 floats: keeps float2 stores 8B aligned,
                  // and 18*m mod 64 distinct for m=0..15 (no bank conflicts)

#define F_RELU 1
#define F_ACC  2

// ---------------------------------------------------------------------
// Generic fp32 WMMA GEMM:  C[M x Nc] (+)= gather(A)[M x K] @ W[K x Nc] (+bias)
// gidx == nullptr -> identity row mapping. flags: F_RELU, F_ACC.
// M multiple of 32, Nc multiple of 64 (true for all our calls).
// FULLK: K is a multiple of TB_K (no tail guards in the hot loop).
// ---------------------------------------------------------------------
template<bool FULLK>
__global__ __launch_bounds__(256)
void wmma_gemm_f32(const float* __restrict__ A, const int* __restrict__ gidx,
                   const float* __restrict__ W, const float* __restrict__ bias,
                   float* __restrict__ C, int M, int K, int Nc, int flags)
{
    __shared__ float As[TB_M][TB_K + A_PAD];
    __shared__ float Bs[TB_K][TB_N];

    const int tid  = threadIdx.x;
    const int lane = tid & 31;
    const int wave = tid >> 5;
    const int mOff = (wave >> 2) * 16;   // 0 or 16
    const int nOff = (wave & 3) * 16;    // 0,16,32,48
    const int blockM = blockIdx.y * TB_M;
    const int blockN = blockIdx.x * TB_N;

    // --- per-thread fixed staging slots (resolved once, registers only) ---
    // A: thread loads float2 at (ar, ak..ak+1); 256 threads cover 32x16.
    const int ar = tid >> 3;            // 0..31
    const int ak = (tid & 7) * 2;       // 0,2,..,14
    const int arow = gidx ? gidx[blockM + ar] : (blockM + ar);
    const float* __restrict__ Aptr = A + (size_t)arow * K + ak;
    // B: thread loads float4 at (bk, bn..bn+3); 256 threads cover 16x64.
    const int bk = tid >> 4;            // 0..15
    const int bn = (tid & 15) * 4;
    const float* __restrict__ Wptr = W + (size_t)bk * Nc + blockN + bn;

    const int halfSel = lane >> 4;      // 0: lanes 0-15, 1: lanes 16-31
    const int mlane   = lane & 15;

    v8f acc = {};

    for (int k0 = 0; k0 < K; k0 += TB_K) {
        // ---- stage A tile (32x16): one b64 load + one b64 LDS store ----
        float2 av;
        if (FULLK) {
            av = *(const float2*)(Aptr + k0);
        } else {
            av.x = (k0 + ak + 0 < K) ? Aptr[k0 + 0] : 0.f;
            av.y = (k0 + ak + 1 < K) ? Aptr[k0 + 1] : 0.f;
        }
        *(float2*)&As[ar][ak] = av;

        // ---- stage B tile (16x64): one b128 load + one b128 LDS store ----
        float4 bv;
        if (FULLK || (k0 + bk) < K) {
            bv = *(const float4*)(Wptr + (size_t)k0 * Nc);
        } else {
            bv = make_float4(0.f, 0.f, 0.f, 0.f);
        }
        *(float4*)&Bs[bk][bn] = bv;
        __syncthreads();

        // 4 x V_WMMA_F32_16X16X4_F32 per K-chunk
#pragma unroll
        for (int kk = 0; kk < TB_K; kk += 4) {
            int ka = kk + 2 * halfSel;  // K pair selected by lane half
            v2f a, b;
            a.x = As[mOff + mlane][ka];
            a.y = As[mOff + mlane][ka + 1];
            b.x = Bs[ka][nOff + mlane];
            b.y = Bs[ka + 1][nOff + mlane];
            acc = __builtin_amdgcn_wmma_f32_16x16x4_f32(
                false, a, false, b, (short)0, acc, false, false);
        }
        __syncthreads();
    }

    // ---- store 16x16 tile: VGPR v holds M = v (+8 for upper lane half) ----
#pragma unroll
    for (int v = 0; v < 8; ++v) {
        int row = blockM + mOff + v + (halfSel << 3);
        int col = blockN + nOff + mlane;
        size_t o = (size_t)row * Nc + col;
        float val = acc[v];
        if (bias) val += bias[col];
        if (flags & F_ACC) val += C[o];
        if (flags & F_RELU) val = fmaxf(val, 0.f);
        C[o] = val;
    }
}

// effective fusion bias: fb[c] + sum_k fusion_w[k][c] (the all-ones half)
__global__ void fusion_bias_kernel(const float* __restrict__ fw,
                                   const float* __restrict__ fb,
                                   float* __restrict__ biaseff)
{
    int c = threadIdx.x;
    float s = fb[c];
    for (int k = 0; k < HDIM; ++k) s += fw[k * HDIM + c];
    biaseff[c] = s;
}

// e_hat = Ce + Dx[dst] + Ex[src]; sigma = sigmoid(e_hat);
// num[dst] += sigma * Bx[src]; den[dst] += sigma   (float atomics, L2-resident)
__global__ __launch_bounds__(256)
void edge_gate_kernel(const int* __restrict__ src, const int* __restrict__ dst,
                      const float* __restrict__ Bx, const float* __restrict__ Dx,
                      const float* __restrict__ Ex, float* __restrict__ ehat,
                      float* __restrict__ numB, float* __restrict__ denB)
{
    long long t = (long long)blockIdx.x * blockDim.x + threadIdx.x;
    const long long total = (long long)EDGES * (HDIM / 4);
    if (t >= total) return;
    int eid = (int)(t >> 6);           // HDIM/4 == 64
    int c   = (int)(t & 63) << 2;
    int s = src[eid], d = dst[eid];
    size_t eo = (size_t)eid * HDIM + c;
    size_t so = (size_t)s   * HDIM + c;
    size_t go = (size_t)d   * HDIM + c;

    float4 ce = *(const float4*)(ehat + eo);
    float4 dx = *(const float4*)(Dx + go);
    float4 ex = *(const float4*)(Ex + so);
    float4 bx = *(const float4*)(Bx + so);

    float4 eh;
    eh.x = ce.x + dx.x + ex.x;
    eh.y = ce.y + dx.y + ex.y;
    eh.z = ce.z + dx.z + ex.z;
    eh.w = ce.w + dx.w + ex.w;
    *(float4*)(ehat + eo) = eh;

    float sg;
    sg = 1.f / (1.f + __expf(-eh.x)); atomicAdd(denB + go + 0, sg); atomicAdd(numB + go + 0, sg * bx.x);
    sg = 1.f / (1.f + __expf(-eh.y)); atomicAdd(denB + go + 1, sg); atomicAdd(numB + go + 1, sg * bx.y);
    sg = 1.f / (1.f + __expf(-eh.z)); atomicAdd(denB + go + 2, sg); atomicAdd(numB + go + 2, sg * bx.z);
    sg = 1.f / (1.f + __expf(-eh.w)); atomicAdd(denB + go + 3, sg); atomicAdd(numB + go + 3, sg * bx.w);
}

// per-column (training BN) partial sums; thread = column (coalesced).
// If numB != null, first does data = data + num/(den+eps) in place (x path).
__global__ __launch_bounds__(256)
void col_stats_kernel(float* __restrict__ data, const float* __restrict__ numB,
                      const float* __restrict__ denB, float* __restrict__ colsum,
                      float* __restrict__ colsq, int rows, int rowsPerBlock)
{
    int c  = threadIdx.x;
    int r0 = blockIdx.x * rowsPerBlock;
    int r1 = r0 + rowsPerBlock; if (r1 > rows) r1 = rows;
    float s = 0.f, ss = 0.f;
    for (int r = r0; r < r1; ++r) {
        size_t i = (size_t)r * HDIM + c;
        float v = data[i];
        if (numB) { v += numB[i] / (denB[i] + AGG_EPS); data[i] = v; }
        s += v; ss += v * v;
    }
    atomicAdd(colsum + c, s);
    atomicAdd(colsq + c, ss);
}

__global__ void bn_finalize_kernel(const float* __restrict__ colsum,
                                   const float* __restrict__ colsq,
                                   float* __restrict__ mean, float* __restrict__ rstd,
                                   float invM)
{
    int c = threadIdx.x;
    float mu  = colsum[c] * invM;
    float var = colsq[c] * invM - mu * mu;
    mean[c] = mu;
    rstd[c] = rsqrtf(var + BN_EPS);
}

// out = xin + relu(g*(xc-mean)*rstd + b), float4 vectorized
__global__ __launch_bounds__(256)
void bn_res_relu_kernel(const float* __restrict__ xin, const float* __restrict__ xc,
                        const float* __restrict__ mean, const float* __restrict__ rstd,
                        const float* __restrict__ g, const float* __restrict__ b,
                        float* __restrict__ xout, long long rows)
{
    long long t = (long long)blockIdx.x * blockDim.x + threadIdx.x;
    long long total4 = rows * (HDIM / 4);
    if (t >= total4) return;
    int c = (int)(t & 63) << 2;
    size_t o = (size_t)t * 4;
    float4 v  = *(const float4*)(xc + o);
    float4 in = *(const float4*)(xin + o);
    float4 r;
    r.x = in.x + fmaxf(g[c + 0] * (v.x - mean[c + 0]) * rstd[c + 0] + b[c + 0], 0.f);
    r.y = in.y + fmaxf(g[c + 1] * (v.y - mean[c + 1]) * rstd[c + 1] + b[c + 1], 0.f);
    r.z = in.z + fmaxf(g[c + 2] * (v.z - mean[c + 2]) * rstd[c + 2] + b[c + 2], 0.f);
    r.w = in.w + fmaxf(g[c + 3] * (v.w - mean[c + 3]) * rstd[c + 3] + b[c + 3], 0.f);
    *(float4*)(xout + o) = r;
}

// out[e] = dot(h[e,:], w2) + b2 ; one wave32 per edge, shuffle reduce
__global__ __launch_bounds__(256)
void dec2_kernel(const float* __restrict__ h, const float* __restrict__ w2,
                 const float* __restrict__ b2, float* __restrict__ out, int rows)
{
    int lane = threadIdx.x & 31;
    int wid  = blockIdx.x * (blockDim.x >> 5) + (threadIdx.x >> 5);
    if (wid >= rows) return;
    const float* row = h + (size_t)wid * HDIM;
    float s = 0.f;
#pragma unroll
    for (int j = 0; j < HDIM / 32; ++j) s += row[lane + 32 * j] * w2[lane + 32 * j];
#pragma unroll
    for (int off = 16; off > 0; off >>= 1) s += __shfl_xor(s, off, 32);
    if (lane == 0) out[wid] = s + b2[0];
}

// ---------------------------------------------------------------------
extern "C" void kernel_launch(void* const* d_in, const int* in_sizes, int n_in,
                              void* d_out, int out_size, void* d_ws, size_t ws_size,
                              hipStream_t stream)
{
    const int*   ei       = (const int*)d_in[0];
    const float* aligned  = (const float*)d_in[1];
    const float* h_old    = (const float*)d_in[2];
    const float* fusion_w = (const float*)d_in[4];
    const float* fusion_b = (const float*)d_in[5];
    const float* eproj_w  = (const float*)d_in[6];
    const float* eproj_b  = (const float*)d_in[7];
    const float* A_w = (const float*)d_in[8];  const float* A_b = (const float*)d_in[9];
    const float* B_w = (const float*)d_in[10]; const float* B_b = (const float*)d_in[11];
    const float* C_w = (const float*)d_in[12]; const float* C_b = (const float*)d_in[13];
    const float* D_w = (const float*)d_in[14]; const float* D_b = (const float*)d_in[15];
    const float* E_w = (const float*)d_in[16]; const float* E_b = (const float*)d_in[17];
    const float* bn_x_g = (const float*)d_in[18]; const float* bn_x_b = (const float*)d_in[19];
    const float* bn_e_g = (const float*)d_in[20]; const float* bn_e_b = (const float*)d_in[21];
    const float* dec1_w = (const float*)d_in[22]; const float* dec1_b = (const float*)d_in[23];
    const float* dec2_w = (const float*)d_in[24]; const float* dec2_b = (const float*)d_in[25];

    const int* srcI = ei;
    const int* dstI = ei + EDGES;

    const size_t NH = (size_t)NNODES * HDIM;   // 25.6M floats
    const size_t EH = (size_t)EDGES  * HDIM;   // 76.8M floats

    float* ws   = (float*)d_ws;
    float* xA   = ws;
    float* xB   = ws + NH;
    float* bAx  = ws + 2 * NH;
    float* bBx  = ws + 3 * NH;
    float* bDx  = ws + 4 * NH;
    float* bEx  = ws + 5 * NH;
    float* bNum = ws + 6 * NH;
    float* bDen = ws + 7 * NH;
    float* eA   = ws + 8 * NH;
    float* eB   = eA + EH;
    float* eH   = eA + 2 * EH;        // Ce -> e_hat -> decoder hidden
    float* colsum  = eA + 3 * EH;     // [H]
    float* colsq   = colsum + HDIM;   // [H]
    float* meanB   = colsum + 2 * HDIM;
    float* rstdB   = colsum + 3 * HDIM;
    float* biaseff = colsum + 4 * HDIM;

    dim3 blk(256);
    dim3 gN(HDIM / TB_N, NNODES / TB_M);  // (4, 3125)
    dim3 gE(HDIM / TB_N, EDGES  / TB_M);  // (4, 9375)
    int gEdge = (int)(((long long)EDGES * (HDIM / 4) + 255) / 256);
    int gNH4  = (int)((NH / 4 + 255) / 256);
    int gEH4  = (int)((EH / 4 + 255) / 256);
    int gColN = (NNODES + 255) / 256;
    int gColE = (EDGES + 255) / 256;

    // --- input embedding ---
    fusion_bias_kernel<<<1, 256, 0, stream>>>(fusion_w, fusion_b, biaseff);
    wmma_gemm_f32<true><<<gN, blk, 0, stream>>>(h_old, nullptr, fusion_w + (size_t)HDIM * HDIM,
                                                biaseff, xA, NNODES, HDIM, HDIM, F_RELU);
    wmma_gemm_f32<false><<<gE, blk, 0, stream>>>(aligned, nullptr, eproj_w, eproj_b,
                                                 eA, EDGES, 8, HDIM, 0);

    // --- GatedGCN layers ---
    for (int l = 0; l < LNUM; ++l) {
        size_t wo = (size_t)l * HDIM * HDIM, bo = (size_t)l * HDIM;
        wmma_gemm_f32<true><<<gN, blk, 0, stream>>>(xA, nullptr, A_w + wo, A_b + bo, bAx, NNODES, HDIM, HDIM, 0);
        wmma_gemm_f32<true><<<gN, blk, 0, stream>>>(xA, nullptr, B_w + wo, B_b + bo, bBx, NNODES, HDIM, HDIM, 0);
        wmma_gemm_f32<true><<<gN, blk, 0, stream>>>(xA, nullptr, D_w + wo, D_b + bo, bDx, NNODES, HDIM, HDIM, 0);
        wmma_gemm_f32<true><<<gN, blk, 0, stream>>>(xA, nullptr, E_w + wo, E_b + bo, bEx, NNODES, HDIM, HDIM, 0);
        wmma_gemm_f32<true><<<gE, blk, 0, stream>>>(eA, nullptr, C_w + wo, C_b + bo, eH, EDGES, HDIM, HDIM, 0);

        hipMemsetAsync(bNum, 0, NH * sizeof(float), stream);
        hipMemsetAsync(bDen, 0, NH * sizeof(float), stream);
        edge_gate_kernel<<<gEdge, blk, 0, stream>>>(srcI, dstI, bBx, bDx, bEx, eH, bNum, bDen);

        // node update: xc = Ax + num/(den+eps); BN; relu; residual
        hipMemsetAsync(colsum, 0, 2 * HDIM * sizeof(float), stream);
        col_stats_kernel<<<gColN, blk, 0, stream>>>(bAx, bNum, bDen, colsum, colsq, NNODES, 256);
        bn_finalize_kernel<<<1, 256, 0, stream>>>(colsum, colsq, meanB, rstdB, 1.f / NNODES);
        bn_res_relu_kernel<<<gNH4, blk, 0, stream>>>(xA, bAx, meanB, rstdB,
                                                     bn_x_g + bo, bn_x_b + bo, xB, NNODES);
        { float* t = xA; xA = xB; xB = t; }

        // edge update: BN(e_hat); relu; residual
        hipMemsetAsync(colsum, 0, 2 * HDIM * sizeof(float), stream);
        col_stats_kernel<<<gColE, blk, 0, stream>>>(eH, nullptr, nullptr, colsum, colsq, EDGES, 256);
        bn_finalize_kernel<<<1, 256, 0, stream>>>(colsum, colsq, meanB, rstdB, 1.f / EDGES);
        bn_res_relu_kernel<<<gEH4, blk, 0, stream>>>(eA, eH, meanB, rstdB,
                                                     bn_e_g + bo, bn_e_b + bo, eB, EDGES);
        { float* t = eA; eA = eB; eB = t; }
    }

    // --- decoder: h = relu(x[src]@W1a + x[dst]@W1b + aligned@W1c + b1); out = h@w2 + b2
    wmma_gemm_f32<false><<<gE, blk, 0, stream>>>(aligned, nullptr, dec1_w + (size_t)(2 * HDIM) * HDIM,
                                                 dec1_b, eH, EDGES, 8, HDIM, 0);
    wmma_gemm_f32<true><<<gE, blk, 0, stream>>>(xA, srcI, dec1_w, nullptr,
                                                eH, EDGES, HDIM, HDIM, F_ACC);
    wmma_gemm_f32<true><<<gE, blk, 0, stream>>>(xA, dstI, dec1_w + (size_t)HDIM * HDIM, nullptr,
                                                eH, EDGES, HDIM, HDIM, F_ACC | F_RELU);
    dec2_kernel<<<(EDGES + 7) / 8, blk, 0, stream>>>(eH, dec2_w, dec2_b, (float*)d_out, EDGES);
}